// SageNet_43130061586721
// MI455X (gfx1250) — compile-verified
//
#include <hip/hip_runtime.h>

// ---------------------------------------------------------------------------
// GraphSAGE pipeline for MI455X (gfx1250, wave32).
// Layers 1/2: scatter-mean + tiny fused linear/normalize/relu (VALU).
// Layer 3:    scatter-mean + (80000x36)@(36x216) GEMM via V_WMMA_F32_16X16X4_F32,
//             B matrix staged in LDS (transposed), one wave = 16 rows x all 224 cols.
// Layer 4:    scatter-mean + 216->2 dot products.
// ---------------------------------------------------------------------------

namespace {
constexpr int B    = 16;
constexpr int NTOT = 50000;
constexpr int CIN  = 5;
constexpr int N0   = 20000;
constexpr int N1   = 5000;
constexpr int N2   = 1000;
constexpr int E0   = 320000;
constexpr int E1   = 80000;
constexpr int E2   = 16000;
constexpr int K3   = 36;    // GEMM K  (multiple of 4)
constexpr int NO3  = 216;   // GEMM N  (true)
constexpr int NP3  = 224;   // GEMM N padded to 14*16
constexpr int NTILES = NP3 / 16;          // 14
constexpr int ROWS3  = B * N1;            // 80000
constexpr int MTILES = ROWS3 / 16;        // 5000 (grid 625 blocks x 8 waves, exact)
}

typedef __attribute__((ext_vector_type(2))) float v2f;
typedef __attribute__((ext_vector_type(8))) float v8f;

__global__ void k_zero(float* __restrict__ p, int n) {
    int t = blockIdx.x * blockDim.x + threadIdx.x;
    if (t < n) p[t] = 0.0f;
}

__global__ void k_count(const int* __restrict__ dst, float* __restrict__ cnt, int E) {
    int t = blockIdx.x * blockDim.x + threadIdx.x;
    if (t < E) atomicAdd(&cnt[dst[t]], 1.0f);
}

// Pack W3 (36x216) -> Wt (224x36, TRANSPOSED, zero-padded cols) and b3 -> bp(224).
// Transposed layout makes each WMMA B-fragment K-pair an aligned 8-byte LDS load.
__global__ void k_pack_w3(const float* __restrict__ W3, const float* __restrict__ b3,
                          float* __restrict__ Wt, float* __restrict__ bp) {
    int t = blockIdx.x * blockDim.x + threadIdx.x;
    if (t < NP3 * K3) {
        int c = t / K3, k = t % K3;             // Wt[c][k]
        Wt[t] = (c < NO3) ? W3[k * NO3 + c] : 0.0f;
    }
    if (t < NP3) bp[t] = (t < NO3) ? b3[t] : 0.0f;
}

// xg[b,i,c] = x[b, n_id[i], c]
__global__ void k_gather(const float* __restrict__ x, const int* __restrict__ nid,
                         float* __restrict__ xg, int total) {
    int t = blockIdx.x * blockDim.x + threadIdx.x;
    if (t >= total) return;
    int c = t % CIN;
    int i = (t / CIN) % N0;
    int b = t / (CIN * N0);
    xg[t] = x[((size_t)b * NTOT + nid[i]) * CIN + c];
}

// agg[b,dst,c] += feat[b,src,c] over all edges, all batches, all channels.
__global__ void k_edge_agg(const float* __restrict__ feat, const int* __restrict__ src,
                           const int* __restrict__ dst, float* __restrict__ agg,
                           int Nsrc, int Ndst, int C, int total) {
    int t = blockIdx.x * blockDim.x + threadIdx.x;
    if (t >= total) return;
    int per = B * C;
    int e = t / per;
    int r = t - e * per;
    int b = r / C;
    int c = r - b * C;
    int s = src[e], d = dst[e];
    atomicAdd(&agg[((size_t)b * Ndst + d) * C + c],
              feat[((size_t)b * Nsrc + s) * C + c]);
}

// Layer 1: mean (incl. self loop) -> concat(self, agg) -> @W1+b1 -> l2norm -> relu
__global__ void k_node_l1(const float* __restrict__ xg, const float* __restrict__ agg,
                          const float* __restrict__ cnt, const float* __restrict__ W,
                          const float* __restrict__ bias, float* __restrict__ out,
                          int total) {
    int t = blockIdx.x * blockDim.x + threadIdx.x;  // t = b*N0 + i
    if (t >= total) return;
    int i = t % N0;
    const float* xs = xg  + (size_t)t * CIN;
    const float* ag = agg + (size_t)t * CIN;
    float inv = 1.0f / (cnt[i] + 1.0f);
    float f[10];
#pragma unroll
    for (int k = 0; k < 5; ++k) { f[k] = xs[k]; f[5 + k] = (ag[k] + xs[k]) * inv; }
    float o[6]; float ss = 0.0f;
#pragma unroll
    for (int j = 0; j < 6; ++j) {
        float s = bias[j];
#pragma unroll
        for (int k = 0; k < 10; ++k) s += f[k] * W[k * 6 + j];
        o[j] = s; ss += s * s;
    }
    float invn = 1.0f / fmaxf(sqrtf(ss), 1e-12f);
#pragma unroll
    for (int j = 0; j < 6; ++j) {
        float v = o[j] * invn;
        out[(size_t)t * 6 + j] = v > 0.0f ? v : 0.0f;
    }
}

// Layer 2: same structure, 6 self + 6 agg -> 36, l2norm, relu
__global__ void k_node_l2(const float* __restrict__ h1, const float* __restrict__ agg,
                          const float* __restrict__ cnt, const float* __restrict__ W,
                          const float* __restrict__ bias, float* __restrict__ out,
                          int total) {
    int t = blockIdx.x * blockDim.x + threadIdx.x;
    if (t >= total) return;
    int i = t % N0;
    const float* xs = h1  + (size_t)t * 6;
    const float* ag = agg + (size_t)t * 6;
    float inv = 1.0f / (cnt[i] + 1.0f);
    float f[12];
#pragma unroll
    for (int k = 0; k < 6; ++k) { f[k] = xs[k]; f[6 + k] = (ag[k] + xs[k]) * inv; }
    float o[36]; float ss = 0.0f;
#pragma unroll
    for (int j = 0; j < 36; ++j) {
        float s = bias[j];
#pragma unroll
        for (int k = 0; k < 12; ++k) s += f[k] * W[k * 36 + j];
        o[j] = s; ss += s * s;
    }
    float invn = 1.0f / fmaxf(sqrtf(ss), 1e-12f);
#pragma unroll
    for (int j = 0; j < 36; ++j) {
        float v = o[j] * invn;
        out[(size_t)t * 36 + j] = v > 0.0f ? v : 0.0f;
    }
}

// agg[b,i,c] *= 1/max(cnt[i],1)
__global__ void k_scale(float* __restrict__ agg, const float* __restrict__ cnt,
                        int C, int Nn, int total) {
    int t = blockIdx.x * blockDim.x + threadIdx.x;
    if (t >= total) return;
    int i = (t / C) % Nn;
    agg[t] *= 1.0f / fmaxf(cnt[i], 1.0f);
}

// Layer-3 GEMM: Y(80000x216) = relu(X(80000x36) @ W3 + b3).
// B matrix (transposed, padded: 224x36 = 32KB) + bias staged once per block in LDS.
// Each wave owns one 16-row M-tile and sweeps all 14 N-tiles: A fragment loaded
// once per K-step and reused 14x -> 126 V_WMMA_F32_16X16X4_F32 per wave.
// Grid is exact (625*8 waves = 5000 M-tiles) so EXEC is all-ones for every WMMA.
__global__ void __launch_bounds__(256) k_gemm3(const float* __restrict__ X,
                                               const float* __restrict__ Wt,
                                               const float* __restrict__ bp,
                                               float* __restrict__ Y) {
    __shared__ float sW[NP3 * K3];   // sW[col][k], 8-byte aligned K-pairs
    __shared__ float sb[NP3];
    for (int idx = threadIdx.x; idx < NP3 * K3; idx += 256) sW[idx] = Wt[idx];
    for (int idx = threadIdx.x; idx < NP3; idx += 256) sb[idx] = bp[idx];
    __syncthreads();

    int wave = threadIdx.x >> 5;
    int lane = threadIdx.x & 31;
    int mt   = blockIdx.x * 8 + wave;      // < MTILES by construction
    int half = lane >> 4;                  // selects K pair (A/B) / upper M half (D)
    int n    = lane & 15;                  // A: row-in-tile; B/D: column-in-tile
    int row  = mt * 16 + n;
    const float* xrow = X + (size_t)row * K3;

    v8f acc[NTILES] = {};
#pragma unroll
    for (int kk = 0; kk < K3; kk += 4) {
        v2f a = *(const v2f*)(xrow + kk + 2 * half);       // K-pair, 8B aligned
#pragma unroll
        for (int ntile = 0; ntile < NTILES; ++ntile) {
            int col = ntile * 16 + n;
            v2f b = *(const v2f*)(&sW[col * K3 + kk + 2 * half]);
            acc[ntile] = __builtin_amdgcn_wmma_f32_16x16x4_f32(
                /*neg_a=*/false, a, /*neg_b=*/false, b,
                /*c_mod=*/(short)0, acc[ntile], /*reuse_a=*/false, /*reuse_b=*/false);
        }
    }
#pragma unroll
    for (int ntile = 0; ntile < NTILES; ++ntile) {
        int col = ntile * 16 + n;
        if (col < NO3) {
            float bias = sb[col];
#pragma unroll
            for (int v = 0; v < 8; ++v) {
                int r = mt * 16 + v + 8 * half;   // D layout: M = v (+8 upper half)
                float val = acc[ntile][v] + bias;
                Y[(size_t)r * NO3 + col] = val > 0.0f ? val : 0.0f;
            }
        }
    }
}

// Layer 4: out[b,i,co] = b4[co] + (1/cnt) * sum_k agg3[b,i,k] * W4[k,co]
__global__ void k_out(const float* __restrict__ agg3, const float* __restrict__ cnt,
                      const float* __restrict__ W4, const float* __restrict__ b4,
                      float* __restrict__ out, int total) {
    int t = blockIdx.x * blockDim.x + threadIdx.x;
    if (t >= total) return;
    int co = t & 1;
    int r  = t >> 1;              // r = b*N2 + i
    int i  = r % N2;
    float inv = 1.0f / fmaxf(cnt[i], 1.0f);
    const float* a = agg3 + (size_t)r * NO3;
    float s = 0.0f;
    for (int k = 0; k < NO3; ++k) s += a[k] * W4[k * 2 + co];
    out[t] = b4[co] + s * inv;
}

extern "C" void kernel_launch(void* const* d_in, const int* in_sizes, int n_in,
                              void* d_out, int out_size, void* d_ws, size_t ws_size,
                              hipStream_t stream) {
    (void)in_sizes; (void)n_in; (void)out_size; (void)ws_size;
    const float* x   = (const float*)d_in[0];
    const int*   nid = (const int*)d_in[1];
    const int*   ei0 = (const int*)d_in[2];   // [src(E0); dst(E0)]
    const int*   ei1 = (const int*)d_in[3];
    const int*   ei2 = (const int*)d_in[4];
    const float* W1  = (const float*)d_in[5];
    const float* b1  = (const float*)d_in[6];
    const float* W2  = (const float*)d_in[7];
    const float* b2  = (const float*)d_in[8];
    const float* W3  = (const float*)d_in[9];
    const float* b3  = (const float*)d_in[10];
    const float* W4  = (const float*)d_in[11];
    const float* b4  = (const float*)d_in[12];
    float* ws = (float*)d_ws;

    // Workspace layout (floats), slots reused across phases.
    float* cnt0 = ws + 0;                    // 20000
    float* cnt1 = ws + 20000;                // 5000
    float* cnt2 = ws + 25000;                // 1000
    float* Wt   = ws + 26000;                // 224*36 = 8064 (transposed, padded W3)
    float* bp   = ws + 26000 + NP3 * K3;     // 224
    size_t off  = 34304;
    float* S1 = ws + off; off += 2880000;    // xg (1.6M) -> agg2 (2.88M)
    float* S2 = ws + off; off += 3456000;    // agg0 -> agg1 -> agg3
    float* S3 = ws + off; off += 1920000;    // h1
    float* S4 = ws + off; off += 11520000;   // h2
    float* S5 = ws + off; off += 17280000;   // h3

    auto g = [](int n) { return (n + 255) / 256; };

    // Counters (cnt0,cnt1,cnt2 are contiguous) + degree counts.
    k_zero<<<g(26000), 256, 0, stream>>>(cnt0, 26000);
    k_count<<<g(E0), 256, 0, stream>>>(ei0 + E0, cnt0, E0);
    k_count<<<g(E1), 256, 0, stream>>>(ei1 + E1, cnt1, E1);
    k_count<<<g(E2), 256, 0, stream>>>(ei2 + E2, cnt2, E2);
    k_pack_w3<<<g(NP3 * K3), 256, 0, stream>>>(W3, b3, Wt, bp);

    // Gather x[:, n_id, :]
    k_gather<<<g(B * N0 * CIN), 256, 0, stream>>>(x, nid, S1, B * N0 * CIN);

    // Layer 1
    k_zero<<<g(B * N0 * CIN), 256, 0, stream>>>(S2, B * N0 * CIN);
    k_edge_agg<<<g(E0 * B * CIN), 256, 0, stream>>>(S1, ei0, ei0 + E0, S2, N0, N0, CIN, E0 * B * CIN);
    k_node_l1<<<g(B * N0), 256, 0, stream>>>(S1, S2, cnt0, W1, b1, S3, B * N0);

    // Layer 2
    k_zero<<<g(B * N0 * 6), 256, 0, stream>>>(S2, B * N0 * 6);
    k_edge_agg<<<g(E0 * B * 6), 256, 0, stream>>>(S3, ei0, ei0 + E0, S2, N0, N0, 6, E0 * B * 6);
    k_node_l2<<<g(B * N0), 256, 0, stream>>>(S3, S2, cnt0, W2, b2, S4, B * N0);

    // Layer 3: scatter-mean then WMMA GEMM (LDS-resident B, bias+relu fused)
    k_zero<<<g(B * N1 * K3), 256, 0, stream>>>(S1, B * N1 * K3);
    k_edge_agg<<<g(E1 * B * K3), 256, 0, stream>>>(S4, ei1, ei1 + E1, S1, N0, N1, K3, E1 * B * K3);
    k_scale<<<g(B * N1 * K3), 256, 0, stream>>>(S1, cnt1, K3, N1, B * N1 * K3);
    k_gemm3<<<MTILES / 8, 256, 0, stream>>>(S1, Wt, bp, S5);  // 625 blocks, exact

    // Layer 4
    k_zero<<<g(B * N2 * NO3), 256, 0, stream>>>(S2, B * N2 * NO3);
    k_edge_agg<<<g(E2 * B * NO3), 256, 0, stream>>>(S5, ei2, ei2 + E2, S2, N1, N2, NO3, E2 * B * NO3);
    k_out<<<g(B * N2 * 2), 256, 0, stream>>>(S2, cnt2, W4, b4, (float*)d_out, B * N2 * 2);
}